// RoIRotate_37649683316743
// MI455X (gfx1250) — compile-verified
//
#include <hip/hip_runtime.h>
#include <hip/hip_bf16.h>
#include <math.h>

// RoIRotate (theta == 0 always) for MI455X / gfx1250.
//
// Bandwidth-bound kernel (~230 MB DRAM traffic vs 0.2 GFLOP): image (128 MB)
// is L2-resident (192 MB L2), output is 101 MB of stores. Strategy:
//  Stage 1: separable horizontal interpolation -> LDS  (2 gathers/elem)
//  Stage 2: vertical interpolation as Wy(16x32) x HR(32x16) per channel on the
//           matrix pipe via V_WMMA_F32_16X16X4_F32 (fp32, K=32 in 8 steps)
// One block per (batch, box, 16-column tile); 256 threads = 8 wave32.

typedef float v2f __attribute__((ext_vector_type(2)));
typedef float v8f __attribute__((ext_vector_type(8)));

constexpr int Bc   = 4;
constexpr int Cc   = 32;
constexpr int Hc   = 512;
constexpr int Wc   = 512;
constexpr int Nc   = 64;
constexpr int HOUT = 16;
constexpr int MAXW = 192;
constexpr int JT   = MAXW / 16;   // 12 column tiles

__global__ __launch_bounds__(256)
void RoIRotate_37649683316743_kernel(const float* __restrict__ img,
                                     const float* __restrict__ boxes,
                                     float* __restrict__ out,
                                     float* __restrict__ outMask) {
  // hr[rowslot k (0..31)][channel c (0..31)][col j (0..15)]  == 64 KB LDS
  __shared__ float hr[32 * Cc * 16];

  const int jtile = blockIdx.x;
  const int n     = blockIdx.y;
  const int b     = blockIdx.z;
  const int tid   = threadIdx.x;

  // Box parameters (uniform per block; recompute per thread, it's cheap).
  const float* bx  = boxes + ((size_t)b * Nc + n) * 5;
  const float left = bx[0];
  const float top  = bx[1];
  const float bw   = bx[2] - left;
  const float bh   = bx[3] - top;
  const int   width  = (int)(bw / bh * (float)HOUT);       // trunc == astype(int32)
  const float each_w = bw / (float)(width - 1);
  const float each_h = bh / (float)(HOUT - 1);

  // ---------------- Stage 1: horizontal interpolation into LDS -------------
  // Thread mapping: j = tid&15 (constant per thread -> hoist x math),
  // cbase = tid>>4, loop over 32 row slots x 2 channel halves.
  const int   j   = tid & 15;
  const int   jj  = jtile * 16 + j;
  const float xf  = left + (float)jj * each_w;
  const int   xfi = (int)floorf(xf);
  const int   x0  = min(max(xfi, 0), Wc - 1);
  const int   x1  = min(max(xfi + 1, 0), Wc - 1);
  const float wx0 = (float)x1 - xf;                        // matches clipped-weight ref
  const float wx1 = xf - (float)x0;
  const int   cbase = tid >> 4;                            // 0..15

  for (int k = 0; k < 32; ++k) {
    const int   i   = k >> 1;                              // output row
    const float yf  = top + (float)i * each_h;
    const int   yfi = (int)floorf(yf);
    const int   row = min(max(yfi + (k & 1), 0), Hc - 1);  // tap 0 / tap 1 row
    // Prefetch the second channel-half's line while we chew on the first.
    __builtin_prefetch(img + (((size_t)b * Cc + cbase + 16) * Hc + row) * Wc + x0, 0, 0);
#pragma unroll
    for (int cc = 0; cc < 2; ++cc) {
      const int    c = cbase + (cc << 4);
      const float* p = img + (((size_t)b * Cc + c) * Hc + row) * Wc;
      hr[(k * Cc + c) * 16 + j] = wx0 * p[x0] + wx1 * p[x1];
    }
  }
  __syncthreads();

  // ---------------- Stage 2: vertical interpolation via WMMA ---------------
  // Out(16 rows i x 16 cols j) = Wy(16 x 32 rowslots) * HR(32 x 16), per channel.
  // Wy[i, 2i] = wy0(i), Wy[i, 2i+1] = wy1(i), zeros elsewhere.
  const int wave   = tid >> 5;     // 0..7, owns channels wave*4 .. wave*4+3
  const int lane   = tid & 31;
  const int M      = lane & 15;    // A-matrix row / D column index
  const int hiHalf = lane >> 4;    // lane half selects K pair (A) / row half (D)

  // Per-lane vertical weights for output row M (with reference's clip rule).
  const float yfm = top + (float)M * each_h;
  const int   ymi = (int)floorf(yfm);
  const int   ym0 = min(max(ymi, 0), Hc - 1);
  const int   ym1 = min(max(ymi + 1, 0), Hc - 1);
  const float wy0 = (float)ym1 - yfm;
  const float wy1 = yfm - (float)ym0;

  // A fragments for the 8 K-chunks (16x4 f32: lane<16 holds K={4k,4k+1},
  // lane>=16 holds K={4k+2,4k+3}; VGPR0=.x=even K, VGPR1=.y=odd K).
  v2f afrag[8];
#pragma unroll
  for (int kk = 0; kk < 8; ++kk) {
    const int  kb  = 4 * kk + 2 * hiHalf;     // even K index this lane holds
    const bool hit = ((kb >> 1) == M);        // Wy row M has nonzeros at K=2M,2M+1
    afrag[kk].x = hit ? wy0 : 0.0f;
    afrag[kk].y = hit ? wy1 : 0.0f;
  }

  const float colmask = (jj < width) ? 1.0f : 0.0f;

#pragma unroll 1
  for (int c = wave * 4; c < wave * 4 + 4; ++c) {
    v8f acc = {};
#pragma unroll
    for (int kk = 0; kk < 8; ++kk) {
      const int kb  = 4 * kk + 2 * hiHalf;
      const int idx = (kb * Cc + c) * 16 + M;   // B 4x16: lane half selects K pair
      v2f bfrag;
      bfrag.x = hr[idx];                        // K = kb,   N = M (this lane's col)
      bfrag.y = hr[idx + Cc * 16];              // K = kb+1
      acc = __builtin_amdgcn_wmma_f32_16x16x4_f32(
          /*neg_a=*/false, afrag[kk], /*neg_b=*/false, bfrag,
          /*c_mod=*/(short)0, acc, /*reuse_a=*/false, /*reuse_b=*/false);
    }
    // D layout: lanes 0-15 hold rows 0..7 in VGPRs 0..7, lanes 16-31 rows 8..15.
    const int rowbase = hiHalf * 8;
    float* o = out + ((((size_t)b * Nc + n) * Cc + c) * HOUT + rowbase) * MAXW + jj;
#pragma unroll
    for (int v = 0; v < 8; ++v) {
      o[v * MAXW] = acc[v] * colmask;           // 16 lanes -> 64B coalesced per row
    }
  }

  // Mask output (stored as float 0/1 after the result block).
  if (tid < 16) {
    outMask[((size_t)b * Nc + n) * MAXW + jj] = colmask;
  }
}

extern "C" void kernel_launch(void* const* d_in, const int* in_sizes, int n_in,
                              void* d_out, int out_size, void* d_ws, size_t ws_size,
                              hipStream_t stream) {
  const float* img   = (const float*)d_in[0];   // (4, 32, 512, 512) fp32
  const float* boxes = (const float*)d_in[1];   // (4, 64, 5) fp32
  float* out = (float*)d_out;
  const size_t RES_ELEMS = (size_t)Bc * Nc * Cc * HOUT * MAXW;  // 25,165,824
  float* mask = out + RES_ELEMS;

  dim3 grid(JT, Nc, Bc);   // 12 x 64 x 4 = 3072 blocks
  RoIRotate_37649683316743_kernel<<<grid, 256, 0, stream>>>(img, boxes, out, mask);
}